// MultiGraphAttention_43473658970533
// MI455X (gfx1250) — compile-verified
//
#include <hip/hip_runtime.h>
#include <hip/hip_bf16.h>

typedef __attribute__((ext_vector_type(2))) float v2f;
typedef __attribute__((ext_vector_type(8))) float v8f;

#define DIN   128
#define DOUT  64
#define NHEAD 4
#define DTOT  (NHEAD * DOUT)   // 256
#define LDS_PITCH 132          // feature tile pitch: 128+4 -> conflict-free A reads
#define W_PITCH   72           // W row pitch: 2*72 % 64 == 16 -> halves hit disjoint banks
#define TILES_PER_BLOCK 4      // node tiles per block; amortizes W staging 4x

// ---------- order-preserving float<->uint encoding for atomicMax ----------
__device__ __forceinline__ unsigned ordf(float f) {
    int i = __float_as_int(f);
    return (unsigned)(i >= 0 ? (i | 0x80000000) : ~i);
}
__device__ __forceinline__ float deordf(unsigned u) {
    int i = (u & 0x80000000u) ? (int)(u & 0x7FFFFFFFu) : ~(int)u;
    return __int_as_float(i);
}

// =====================================================================
// Kernel 0: zero the output accumulator and per-node softmax state.
// =====================================================================
__global__ void gat_init(float* __restrict__ out, unsigned* __restrict__ mx,
                         float* __restrict__ den, int total, int nh) {
    int stride = gridDim.x * blockDim.x;
    for (int i = blockIdx.x * blockDim.x + threadIdx.x; i < total; i += stride)
        out[i] = 0.0f;
    for (int i = blockIdx.x * blockDim.x + threadIdx.x; i < nh; i += stride) {
        mx[i]  = 0u;     // below every ordf() encoding -> acts as -inf
        den[i] = 0.0f;
    }
}

// =====================================================================
// Kernel 1: per-head projection x = features @ W_h via f32 WMMA,
// fused with s1 = x.a1, s2 = x.a2 (16-lane shuffle reduction).
// Block = 128 threads = 4 waves; wave w == head w.
// All of W staged in LDS once per block; 4 node tiles per block reuse it.
// Inner loop is LDS-only: ds_load_2addr_b64 (A) + ds_load_2addr_b32 (B) + WMMA.
// =====================================================================
__global__ __launch_bounds__(128)
void gat_project(const float* __restrict__ feat, const float* __restrict__ W,
                 const float* __restrict__ a1g, const float* __restrict__ a2g,
                 float* __restrict__ x, float* __restrict__ s1, float* __restrict__ s2,
                 int nNodes) {
    __shared__ float wlds[NHEAD * DIN * W_PITCH];   // 4*128*72 floats = 144 KB
    __shared__ float atile[16 * LDS_PITCH];         // 8.25 KB

    const int tid  = threadIdx.x;
    const int lane = tid & 31;
    const int h    = tid >> 5;       // wave id == head
    const int l16  = lane & 15;
    const int hl   = lane >> 4;      // half-wave: 0 -> K pair (k0,k0+1), 1 -> (k0+2,k0+3)

    // ---- stage W[4][128][64] into LDS with row pitch 72 (coalesced float4) ----
    for (int i = tid; i < NHEAD * DIN * (DOUT / 4); i += 128) {
        const int rr = i >> 4;                // flat (h,k) row, 0..511
        const int j  = (i & 15) << 2;         // column 0..60 step 4
        float4 v = *(const float4*)(W + (size_t)rr * DOUT + j);
        float* dst = &wlds[rr * W_PITCH + j];
        dst[0] = v.x; dst[1] = v.y; dst[2] = v.z; dst[3] = v.w;
    }

    // attention vectors for this head (per-lane slice across 4 N-tiles)
    float a1v[4], a2v[4];
    #pragma unroll
    for (int nt = 0; nt < 4; ++nt) {
        a1v[nt] = a1g[h * DOUT + nt * 16 + l16];
        a2v[nt] = a2g[h * DOUT + nt * 16 + l16];
    }

    const float* whl  = &wlds[h * DIN * W_PITCH];
    const float* arow = &atile[l16 * LDS_PITCH + 2 * hl];

    for (int t = 0; t < TILES_PER_BLOCK; ++t) {
        const int node0 = (blockIdx.x * TILES_PER_BLOCK + t) * 16;
        if (node0 >= nNodes) break;            // uniform per block

        __syncthreads();   // t==0: W staged; t>0: previous tile done with atile

        // ---- stage 16x128 feature tile into LDS (coalesced float4) ----
        for (int i = tid; i < 16 * DIN / 4; i += 128) {
            const int r = i >> 5;
            const int k = (i & 31) << 2;
            const int node = node0 + r;
            float4 v = make_float4(0.f, 0.f, 0.f, 0.f);
            if (node < nNodes)
                v = *(const float4*)(feat + (size_t)node * DIN + k);
            float* dst = &atile[r * LDS_PITCH + k];
            dst[0] = v.x; dst[1] = v.y; dst[2] = v.z; dst[3] = v.w;
        }
        __syncthreads();

        v8f c0 = {}, c1 = {}, c2 = {}, c3 = {};

        // ---- K loop: 32 steps of V_WMMA_F32_16X16X4_F32 per N-tile ----
        for (int k0 = 0; k0 < DIN; k0 += 4) {
            v2f a;               // A 16x4: lanes 0-15 hold K=k0,k0+1; 16-31 hold k0+2,k0+3
            a.x = arow[k0];
            a.y = arow[k0 + 1];
            const float* wbase = whl + (k0 + 2 * hl) * W_PITCH + l16;
            #pragma unroll
            for (int nt = 0; nt < 4; ++nt) {
                v2f b;           // B 4x16, mirrored layout; pair 256+32B apart -> 2addr_b32
                b.x = wbase[nt * 16];
                b.y = wbase[nt * 16 + W_PITCH];
                v8f* cc = (nt == 0) ? &c0 : (nt == 1) ? &c1 : (nt == 2) ? &c2 : &c3;
                *cc = __builtin_amdgcn_wmma_f32_16x16x4_f32(
                          false, a, false, b, (short)0, *cc, false, false);
            }
        }

        // ---- epilogue: store x[node][h*64+o], reduce s1/s2 ----
        v8f cacc[4] = { c0, c1, c2, c3 };
        #pragma unroll
        for (int r = 0; r < 8; ++r) {
            const int node = node0 + r + 8 * hl;   // C layout: lanes 0-15 M=r, 16-31 M=r+8
            float p1 = 0.f, p2 = 0.f;
            #pragma unroll
            for (int nt = 0; nt < 4; ++nt) {
                float v = cacc[nt][r];
                if (node < nNodes)
                    x[(size_t)node * DTOT + h * DOUT + nt * 16 + l16] = v;
                p1 += v * a1v[nt];
                p2 += v * a2v[nt];
            }
            #pragma unroll
            for (int off = 8; off >= 1; off >>= 1) {
                p1 += __shfl_xor(p1, off, 16);
                p2 += __shfl_xor(p2, off, 16);
            }
            if (l16 == 0 && node < nNodes) {
                s1[node * NHEAD + h] = p1;
                s2[node * NHEAD + h] = p2;
            }
        }
    }
}

// =====================================================================
// Kernel 2: edge logits + segment max (per destination row).
// =====================================================================
__global__ void gat_edge_max(const float* __restrict__ s1, const float* __restrict__ s2,
                             const int* __restrict__ row, const int* __restrict__ col,
                             unsigned* __restrict__ mx, int nEdges) {
    int e = blockIdx.x * blockDim.x + threadIdx.x;
    if (e >= nEdges) return;
    const int r = row[e], c = col[e];
    const float4 s1v = *(const float4*)(s1 + (size_t)r * NHEAD);
    const float4 s2v = *(const float4*)(s2 + (size_t)c * NHEAD);
    const float* p1 = (const float*)&s1v;
    const float* p2 = (const float*)&s2v;
    #pragma unroll
    for (int h = 0; h < NHEAD; ++h) {
        float s  = p1[h] + p2[h];
        float ev = s > 0.f ? s : 0.2f * s;          // leaky_relu(0.2)
        atomicMax(&mx[r * NHEAD + h], ordf(ev));
    }
}

// =====================================================================
// Kernel 3: exp(e - max) per edge, accumulate softmax denominators.
// =====================================================================
__global__ void gat_edge_exp(const float* __restrict__ s1, const float* __restrict__ s2,
                             const int* __restrict__ row, const int* __restrict__ col,
                             const unsigned* __restrict__ mx, float* __restrict__ alpha,
                             float* __restrict__ den, int nEdges) {
    int e = blockIdx.x * blockDim.x + threadIdx.x;
    if (e >= nEdges) return;
    const int r = row[e], c = col[e];
    const float4 s1v = *(const float4*)(s1 + (size_t)r * NHEAD);
    const float4 s2v = *(const float4*)(s2 + (size_t)c * NHEAD);
    const float* p1 = (const float*)&s1v;
    const float* p2 = (const float*)&s2v;
    #pragma unroll
    for (int h = 0; h < NHEAD; ++h) {
        float s  = p1[h] + p2[h];
        float ev = s > 0.f ? s : 0.2f * s;
        float m  = deordf(mx[r * NHEAD + h]);
        float ex = __expf(ev - m);
        alpha[(size_t)e * NHEAD + h] = ex;
        atomicAdd(&den[r * NHEAD + h], ex);
    }
}

// =====================================================================
// Kernel 4: SpMM aggregation. One wave per edge; lane k covers 8 of the
// 256 output components. x gather is contiguous per node (L2-resident).
// =====================================================================
__global__ __launch_bounds__(256)
void gat_aggregate(const int* __restrict__ row, const int* __restrict__ col,
                   const float* __restrict__ alpha, const float* __restrict__ den,
                   const float* __restrict__ x, float* __restrict__ out, int nEdges) {
    const int wave = (int)((blockIdx.x * (size_t)blockDim.x + threadIdx.x) >> 5);
    const int lane = threadIdx.x & 31;
    if (wave >= nEdges) return;
    const int r = row[wave], c = col[wave];
    const float4 av = *(const float4*)(alpha + (size_t)wave * NHEAD);
    const float4 dv = *(const float4*)(den + (size_t)r * NHEAD);
    const float w[4] = { av.x / dv.x, av.y / dv.y, av.z / dv.z, av.w / dv.w };
    const float* xs = x   + (size_t)c * DTOT;
    float*       od = out + (size_t)r * DTOT;
    #pragma unroll
    for (int k = 0; k < 8; ++k) {
        const int comp = (k << 5) | lane;          // 0..255, head = k>>1 (compile-time)
        atomicAdd(&od[comp], w[k >> 1] * xs[comp]);
    }
}

// =====================================================================
// Kernel 5: out = relu(out + bias), in place on [N, 256].
// =====================================================================
__global__ void gat_finalize(float* __restrict__ out, const float* __restrict__ bias,
                             int total) {
    int i = blockIdx.x * blockDim.x + threadIdx.x;
    if (i >= total) return;
    float v = out[i] + bias[i & (DTOT - 1)];
    out[i] = v > 0.f ? v : 0.f;
}

// =====================================================================
extern "C" void kernel_launch(void* const* d_in, const int* in_sizes, int n_in,
                              void* d_out, int out_size, void* d_ws, size_t ws_size,
                              hipStream_t stream) {
    const float* feat = (const float*)d_in[0];
    const float* W    = (const float*)d_in[1];
    const float* a1   = (const float*)d_in[2];
    const float* a2   = (const float*)d_in[3];
    const float* bias = (const float*)d_in[4];
    const int*   row  = (const int*)d_in[5];
    const int*   col  = (const int*)d_in[6];
    float*       out  = (float*)d_out;

    const int nNodes = in_sizes[0] / DIN;     // 50000
    const int nEdges = in_sizes[5];           // 800000

    // workspace layout (all 16-byte aligned slices)
    float*    x     = (float*)d_ws;                             // [N][256]
    float*    s1    = x   + (size_t)nNodes * DTOT;              // [N][4]
    float*    s2    = s1  + (size_t)nNodes * NHEAD;             // [N][4]
    unsigned* mx    = (unsigned*)(s2 + (size_t)nNodes * NHEAD); // [N][4]
    float*    den   = (float*)(mx + (size_t)nNodes * NHEAD);    // [N][4]
    float*    alpha = den + (size_t)nNodes * NHEAD;             // [E][4]

    const int totalOut = nNodes * DTOT;
    const int nh       = nNodes * NHEAD;
    const int nodesPerBlock = 16 * TILES_PER_BLOCK;

    gat_init<<<(totalOut + 255) / 256, 256, 0, stream>>>(out, mx, den, totalOut, nh);

    gat_project<<<(nNodes + nodesPerBlock - 1) / nodesPerBlock, 128, 0, stream>>>(
        feat, W, a1, a2, x, s1, s2, nNodes);

    gat_edge_max<<<(nEdges + 255) / 256, 256, 0, stream>>>(s1, s2, row, col, mx, nEdges);

    gat_edge_exp<<<(nEdges + 255) / 256, 256, 0, stream>>>(s1, s2, row, col, mx, alpha, den, nEdges);

    gat_aggregate<<<(nEdges + 7) / 8, 256, 0, stream>>>(row, col, alpha, den, x, out, nEdges);

    gat_finalize<<<(totalOut + 255) / 256, 256, 0, stream>>>(out, bias, totalOut);
}